// LmHead_32177894982343
// MI455X (gfx1250) — compile-verified
//
#include <hip/hip_runtime.h>

typedef __attribute__((ext_vector_type(2))) float v2f;
typedef __attribute__((ext_vector_type(4))) float v4f;
typedef __attribute__((ext_vector_type(8))) float v8f;

#define EPSF 1e-6f

// ---------------------------------------------------------------------------
// Kernel 1: RMSNorm of 8 rows of length D.  h[b][d] = x*rsqrt(mean(x^2)+eps)*g
// ---------------------------------------------------------------------------
__global__ void rmsnorm_kernel(const float* __restrict__ x,
                               const float* __restrict__ g,
                               float* __restrict__ h, int D) {
  const int row = blockIdx.x;
  const int tid = threadIdx.x;
  const float* xr = x + (size_t)row * D;
  float* hr = h + (size_t)row * D;

  float s = 0.f;
  for (int d = tid; d < D; d += blockDim.x) { float v = xr[d]; s += v * v; }
  // wave32 reduce
  for (int off = 16; off; off >>= 1) s += __shfl_xor(s, off, 32);
  __shared__ float red[32];
  const int wave = tid >> 5, lane = tid & 31;
  if (lane == 0) red[wave] = s;
  __syncthreads();
  __shared__ float sscale;
  if (tid == 0) {
    float tot = 0.f;
    const int nw = blockDim.x >> 5;
    for (int w = 0; w < nw; ++w) tot += red[w];
    sscale = rsqrtf(tot / (float)D + EPSF);
  }
  __syncthreads();
  const float sc = sscale;
  for (int d = tid; d < D; d += blockDim.x) hr[d] = xr[d] * sc * g[d];
}

// ---------------------------------------------------------------------------
// Kernel 2: logits tile GEMV via V_WMMA_F32_16X16X4_F32.
//  - one wave per 16-vocab-row tile, 8 waves per block
//  - h (8 rows) + one zero row staged in LDS (M-padding rows 8..15 read zeros)
//  - weights streamed from HBM with non-temporal float2 loads; each unrolled
//    iteration (K=64) fully consumes its cachelines, so NT eviction is free
//  - dual accumulators break the WMMA RAW chain
//  - per-tile per-batch argmax partials written to workspace
// ---------------------------------------------------------------------------
__global__ void gemv_wmma_kernel(const float* __restrict__ h,   // [8, D]
                                 const float* __restrict__ W,   // [V, D]
                                 float* __restrict__ pvals,     // [8, numTiles]
                                 int*   __restrict__ pidx,      // [8, numTiles]
                                 int D, int V, int numTiles) {
  extern __shared__ float lds[];   // (8+1)*D floats: rows 0..7 = h, row 8 = 0
  const int tid = threadIdx.x;

  // Stage activations into LDS (float4) + build zero pad row.
  {
    v4f* dst = (v4f*)lds;
    const v4f* src = (const v4f*)h;
    const int n4 = (8 * D) >> 2;
    for (int i = tid; i < n4; i += blockDim.x) dst[i] = src[i];
    v4f z = {0.f, 0.f, 0.f, 0.f};
    v4f* zr = (v4f*)(lds + 8 * D);
    for (int i = tid; i < (D >> 2); i += blockDim.x) zr[i] = z;
  }
  __syncthreads();

  const int wave = tid >> 5;
  const int lane = tid & 31;
  const int tile = blockIdx.x * (blockDim.x >> 5) + wave;
  if (tile >= numTiles) return;        // wave-uniform: EXEC stays all-ones
  const int v0 = tile << 4;

  const int col   = lane & 15;         // N for B-matrix, M for A-matrix
  const int khalf = (lane >> 4) << 1;  // K sub-offset: 0 (lanes 0-15) / 2 (16-31)

  const float* wrow = W + (size_t)(v0 + col) * D + khalf;   // B: weight row
  const int arow = (col < 8) ? col : 8;                     // A: pad -> zero row
  const float* als = lds + arow * D + khalf;

  v8f acc0 = {};
  v8f acc1 = {};
  for (int k = 0; k < D; k += 64) {
    v2f a[16], b[16];
#pragma unroll
    for (int c = 0; c < 16; ++c) {
      b[c] = __builtin_nontemporal_load((const v2f*)(wrow + k + 4 * c)); // HBM stream, NT
      a[c] = *(const v2f*)(als + k + 4 * c);                             // ds_load_b64
    }
#pragma unroll
    for (int c = 0; c < 16; c += 2) {
      acc0 = __builtin_amdgcn_wmma_f32_16x16x4_f32(
          false, a[c],     false, b[c],     (short)0, acc0, false, false);
      acc1 = __builtin_amdgcn_wmma_f32_16x16x4_f32(
          false, a[c + 1], false, b[c + 1], (short)0, acc1, false, false);
    }
  }
  v8f acc = acc0 + acc1;

  // D layout: lane L (L<16) holds logits[M=r][N=v0+L] in acc[r], r=0..7 (real batches).
  const int myIdx = v0 + col;
#pragma unroll
  for (int r = 0; r < 8; ++r) {
    float v = acc[r];
    int   i = myIdx;
    // 16-lane tournament argmax, min-index tie-break (lanes 16..31 ignored)
    for (int off = 8; off; off >>= 1) {
      float ov = __shfl_xor(v, off, 16);
      int   oi = __shfl_xor(i, off, 16);
      if (ov > v || (ov == v && oi < i)) { v = ov; i = oi; }
    }
    if (lane == 0) {
      pvals[(size_t)r * numTiles + tile] = v;
      pidx [(size_t)r * numTiles + tile] = i;
    }
  }
}

// ---------------------------------------------------------------------------
// Kernel 3: reduce per-tile partials -> final token per batch row.
// ---------------------------------------------------------------------------
__global__ void final_argmax_kernel(const float* __restrict__ pvals,
                                    const int*   __restrict__ pidx,
                                    float* __restrict__ out, int numTiles) {
  const int b = blockIdx.x;
  const int tid = threadIdx.x;
  float bv = -3.402823466e38f;
  int   bi = 0x7fffffff;
  for (int t = tid; t < numTiles; t += blockDim.x) {
    float v = pvals[(size_t)b * numTiles + t];
    int   i = pidx [(size_t)b * numTiles + t];
    if (v > bv || (v == bv && i < bi)) { bv = v; bi = i; }
  }
  __shared__ float sv[256];
  __shared__ int   si[256];
  sv[tid] = bv; si[tid] = bi;
  __syncthreads();
  for (int s = blockDim.x >> 1; s > 0; s >>= 1) {
    if (tid < s) {
      float ov = sv[tid + s]; int oi = si[tid + s];
      if (ov > sv[tid] || (ov == sv[tid] && oi < si[tid])) { sv[tid] = ov; si[tid] = oi; }
    }
    __syncthreads();
  }
  if (tid == 0) out[b] = (float)si[0];
}

// ---------------------------------------------------------------------------
extern "C" void kernel_launch(void* const* d_in, const int* in_sizes, int n_in,
                              void* d_out, int out_size, void* d_ws, size_t ws_size,
                              hipStream_t stream) {
  const float* hidden = (const float*)d_in[0];
  const float* gamma  = (const float*)d_in[1];
  const float* W      = (const float*)d_in[2];
  const int D = in_sizes[1];              // 4096
  const int V = in_sizes[2] / D;          // 32000
  const int numTiles = (V + 15) / 16;     // 2000

  // workspace layout
  float* h     = (float*)d_ws;                       // 8*D floats
  float* pvals = h + (size_t)8 * D;                  // 8*numTiles floats
  int*   pidx  = (int*)(pvals + (size_t)8 * numTiles);
  (void)n_in; (void)out_size; (void)ws_size;

  rmsnorm_kernel<<<8, 256, 0, stream>>>(hidden, gamma, h, D);

  const int wavesPerBlock = 8;                        // 8 waves x wave32 = 256
  const size_t ldsBytes = (size_t)(8 + 1) * D * sizeof(float);   // 144 KB
  hipFuncSetAttribute((const void*)gemv_wmma_kernel,
                      hipFuncAttributeMaxDynamicSharedMemorySize,
                      (int)ldsBytes);
  dim3 grid((numTiles + wavesPerBlock - 1) / wavesPerBlock);
  gemv_wmma_kernel<<<grid, 32 * wavesPerBlock, ldsBytes, stream>>>(
      h, W, pvals, pidx, D, V, numTiles);

  final_argmax_kernel<<<8, 256, 0, stream>>>(pvals, pidx, (float*)d_out, numTiles);
}